// SpatialGraphConv_6734508720734
// MI455X (gfx1250) — compile-verified
//
#include <hip/hip_runtime.h>

typedef __attribute__((ext_vector_type(2))) float v2f;
typedef __attribute__((ext_vector_type(8))) float v8f;

#define CIN     64
#define COUT    128
#define KNUM    3
#define T_DIM   300
#define V_DIM   25
#define VP      28                 // per-k reduction length padded to mult of 4
#define TSTRIPS 20
#define TS      (T_DIM / TSTRIPS)  // 15 t-slices per workgroup

// LDS row strides (floats) chosen for conflict-free WMMA fragment access
#define WS_ 65   // odd: row-varying-per-lane A-frag reads hit distinct banks
#define XS_ 40   // 2*40 mod 64 == 16: half-wave K-split B-frags on disjoint banks
#define AS_ 40
#define YS_ 34   // conflict-free for D-write halves and stage-2 A-frag reads

__global__ __launch_bounds__(256)
void sgc_fused_wmma(const float* __restrict__ x, const float* __restrict__ A,
                    const float* __restrict__ W, const float* __restrict__ b,
                    float* __restrict__ out) {
  __shared__ float lds_W [KNUM * COUT * WS_]; // 97.5 KB, staged once per WG
  __shared__ float lds_y [COUT * YS_];        // 17.0 KB, per-(t,k) conv result
  __shared__ float lds_x [CIN * XS_];         // 10.0 KB, per-t input slice
  __shared__ float lds_A2[KNUM * VP * AS_];   // 13.1 KB, zero-padded adjacency
  __shared__ float lds_b [KNUM * COUT];       // 1.5 KB

  const int n     = blockIdx.x / TSTRIPS;
  const int strip = blockIdx.x - n * TSTRIPS;
  const int tid  = threadIdx.x;
  const int wave = tid >> 5;        // wave32
  const int lane = tid & 31;
  const int row  = lane & 15;
  const int half = lane >> 4;
  const int kb   = half * 2;        // K-pair base within a 4-chunk

  // ---------------- one-time staging: W, A2 (padded), bias ----------------
  for (int i = tid; i < KNUM * COUT * CIN; i += 256) {
    int r = i >> 6, c = i & 63;                    // CIN == 64
    lds_W[r * WS_ + c] = W[i];
  }
  for (int i = tid; i < KNUM * VP * 32; i += 256) {
    int k  = i / (VP * 32);
    int rm = i - k * (VP * 32);
    int v  = rm >> 5, w = rm & 31;
    lds_A2[(k * VP + v) * AS_ + w] =
        (v < V_DIM && w < V_DIM) ? A[(k * V_DIM + v) * V_DIM + w] : 0.0f;
  }
  for (int i = tid; i < KNUM * COUT; i += 256) lds_b[i] = b[i];

  const float* xb = x   + (size_t)n * CIN  * (T_DIM * V_DIM);
  float*       ob = out + (size_t)n * COUT * (T_DIM * V_DIM);

  for (int it = 0; it < TS; ++it) {
    const int t = strip * TS + it;

    // ---- stage x[n,:,t,:] zero-padded to 32 cols ----
    for (int i = tid; i < CIN * 32; i += 256) {
      int ci = i >> 5, v = i & 31;
      lds_x[ci * XS_ + v] =
          (v < V_DIM) ? xb[(size_t)ci * (T_DIM * V_DIM) + t * V_DIM + v] : 0.0f;
    }
    __syncthreads();   // (first iteration also covers the one-time staging)

    v8f oacc0 = {}, oacc1 = {};    // out accumulators, carried across k
    for (int k = 0; k < KNUM; ++k) {
      // ====== stage 1: y_k[128x25(+pad)] = W_k[128x64] @ x[64x25] + b_k ====
      {
        const int wr = (k * COUT + wave * 16 + row) * WS_;
        v2f af[16];
#pragma unroll
        for (int kc = 0; kc < 16; ++kc) {          // A-frag (W rows), reused x2
          af[kc].x = lds_W[wr + kc * 4 + kb];
          af[kc].y = lds_W[wr + kc * 4 + kb + 1];
        }
#pragma unroll
        for (int nt = 0; nt < 2; ++nt) {
          v8f acc = {};
#pragma unroll
          for (int kc = 0; kc < 16; ++kc) {
            v2f bf;
            bf.x = lds_x[(kc * 4 + kb)     * XS_ + nt * 16 + row];
            bf.y = lds_x[(kc * 4 + kb + 1) * XS_ + nt * 16 + row];
            acc = __builtin_amdgcn_wmma_f32_16x16x4_f32(false, af[kc], false, bf,
                                                        0, acc, false, false);
          }
#pragma unroll
          for (int r = 0; r < 8; ++r) {            // D: M = r + 8*half, N = row
            int cr = wave * 16 + r + half * 8;
            lds_y[cr * YS_ + nt * 16 + row] = acc[r] + lds_b[k * COUT + cr];
          }
        }
      }
      __syncthreads();

      // ====== stage 2: oacc += Y_k[128x28] @ A2_k[28x32] ====================
      // Y_k cols 25..27 hold bias garbage; A2 rows 25..27 are zero, so the
      // padded products vanish — no per-lane guards or divisions needed.
      {
        const int yr = (wave * 16 + row) * YS_;
        v2f yf[7];
#pragma unroll
        for (int kc = 0; kc < 7; ++kc) {           // A-frag from LDS y
          yf[kc].x = lds_y[yr + kc * 4 + kb];
          yf[kc].y = lds_y[yr + kc * 4 + kb + 1];
        }
        const float* a2k = lds_A2 + k * VP * AS_;
#pragma unroll
        for (int kc = 0; kc < 7; ++kc) {
          v2f bf0, bf1;
          bf0.x = a2k[(kc * 4 + kb)     * AS_ + row];
          bf0.y = a2k[(kc * 4 + kb + 1) * AS_ + row];
          bf1.x = a2k[(kc * 4 + kb)     * AS_ + 16 + row];
          bf1.y = a2k[(kc * 4 + kb + 1) * AS_ + 16 + row];
          oacc0 = __builtin_amdgcn_wmma_f32_16x16x4_f32(false, yf[kc], false, bf0,
                                                        0, oacc0, false, false);
          oacc1 = __builtin_amdgcn_wmma_f32_16x16x4_f32(false, yf[kc], false, bf1,
                                                        0, oacc1, false, false);
        }
      }
      __syncthreads();   // protect lds_y (and lds_x on last k) before reuse
    }

    // ---- write out[n, c, t, w] ----
    {
      const int w1 = 16 + row;
#pragma unroll
      for (int r = 0; r < 8; ++r) {
        int cc = wave * 16 + r + half * 8;
        size_t base = (size_t)cc * (T_DIM * V_DIM) + (size_t)t * V_DIM;
        ob[base + row] = oacc0[r];                 // w = 0..15 always valid
        if (w1 < V_DIM) ob[base + w1] = oacc1[r];  // w = 16..24
      }
    }
  }
}

extern "C" void kernel_launch(void* const* d_in, const int* in_sizes, int n_in,
                              void* d_out, int out_size, void* d_ws, size_t ws_size,
                              hipStream_t stream) {
  (void)n_in; (void)out_size; (void)d_ws; (void)ws_size;
  const float* x = (const float*)d_in[0];
  const float* A = (const float*)d_in[1];
  const float* W = (const float*)d_in[2];
  const float* b = (const float*)d_in[3];
  float* out = (float*)d_out;
  const int N = in_sizes[0] / (CIN * T_DIM * V_DIM);   // 64
  dim3 grid((unsigned)(N * TSTRIPS));                  // 1280 persistent blocks
  dim3 block(256);                                     // 8 wave32s
  sgc_fused_wmma<<<grid, block, 0, stream>>>(x, A, W, b, out);
}